// HeteroGNN_11063835754497
// MI455X (gfx1250) — compile-verified
//
#include <hip/hip_runtime.h>

// HeteroGNN on MI455X (gfx1250, wave32).
// All dense layers use V_WMMA_F32_16X16X4_F32 (full fp32: workload is
// L2-gather bound, so fp32 WMMA costs nothing vs bf16 and matches reference
// numerics). Scatter-mean uses hardware global_atomic_add_f32 into
// L2-resident accumulators.

#define H 128
#define LDS_P 34   // padded pitch (floats) for transposed W chunk in LDS

typedef __attribute__((ext_vector_type(2))) float v2f;
typedef __attribute__((ext_vector_type(8))) float v8f;

__device__ __forceinline__ v8f wmma_f32_k4(v2f a, v2f b, v8f c) {
  // D(16x16,f32) = A(16x4,f32) x B(4x16,f32) + C
  return __builtin_amdgcn_wmma_f32_16x16x4_f32(false, a, false, b,
                                               (short)0, c, false, false);
}

// ---------------------------------------------------------------------------
// Generic fused GEMM: out[N x 128] = act( (scale ? diag(scale)*A : A) @ W + b )
// A: [N x K] row-major, W: [K x 128] row-major. Block = 256 thr = 8 waves.
// Each block: 64 rows x 128 cols. Wave w owns col-tile w; 4 row-tiles/wave.
// W staged transposed through LDS in 32-deep K chunks.
// ---------------------------------------------------------------------------
__global__ __launch_bounds__(256) void gemm_bias_act(
    const float* __restrict__ A, const float* __restrict__ W,
    const float* __restrict__ bias, const float* __restrict__ scale,
    float* __restrict__ out, int N, int K, int relu)
{
  __shared__ float ldsWT[H * LDS_P];
  const int tid  = threadIdx.x;
  const int wave = tid >> 5;
  const int lane = tid & 31;
  const int m    = lane & 15;
  const int koff = (lane < 16) ? 0 : 2;   // f32 A/B frag: low half K={0,1}, high half K={2,3}
  const int colBase  = wave * 16;
  const int rowBlock = blockIdx.x * 64;

  const float* arow[4];
  float sc[4];
  #pragma unroll
  for (int rt = 0; rt < 4; ++rt) {
    int r  = rowBlock + rt * 16 + m;
    int rc = (r < N) ? r : (N - 1);       // clamp OOB rows; stores guarded later
    arow[rt] = A + (size_t)rc * K;
    sc[rt]   = scale ? scale[rc] : 1.0f;
  }

  v8f acc[4] = {};
  for (int kc = 0; kc < K; kc += 32) {
    // cooperative transposed stage of W[kc..kc+31][:]
    for (int idx = tid; idx < 32 * H; idx += 256) {
      int kl = idx >> 7;
      int n  = idx & (H - 1);
      float v = (kc + kl < K) ? W[(size_t)(kc + kl) * H + n] : 0.0f;
      ldsWT[n * LDS_P + kl] = v;
    }
    __syncthreads();
    int kchunk = K - kc; if (kchunk > 32) kchunk = 32;
    int ksteps = (kchunk + 3) >> 2;
    for (int ks = 0; ks < ksteps; ++ks) {
      int klocal = ks * 4 + koff;
      v2f b = *(const v2f*)&ldsWT[(colBase + m) * LDS_P + klocal];
      #pragma unroll
      for (int rt = 0; rt < 4; ++rt) {
        v2f a = *(const v2f*)(arow[rt] + kc + klocal);
        a = a * sc[rt];
        acc[rt] = wmma_f32_k4(a, b, acc[rt]);
      }
    }
    __syncthreads();
  }

  // D layout: VGPR v -> M = v (lanes 0-15) / v+8 (lanes 16-31), N = lane%16
  const int ncol = colBase + m;
  const float bb = bias[ncol];
  #pragma unroll
  for (int rt = 0; rt < 4; ++rt) {
    #pragma unroll
    for (int v = 0; v < 8; ++v) {
      int mrow = rowBlock + rt * 16 + v + ((lane < 16) ? 0 : 8);
      float val = acc[rt][v] + bb;
      if (relu) val = fmaxf(val, 0.0f);
      if (mrow < N) out[(size_t)mrow * H + ncol] = val;
    }
  }
}

// ---------------------------------------------------------------------------
// Edge degree counts -> reciprocal (shared by both conv layers)
// ---------------------------------------------------------------------------
__global__ void count_edges(const int* __restrict__ esrc, const int* __restrict__ edst,
                            float* __restrict__ cntL, float* __restrict__ cntT, int E)
{
  int e = blockIdx.x * blockDim.x + threadIdx.x;
  if (e >= E) return;
  unsafeAtomicAdd(cntL + esrc[e], 1.0f);
  unsafeAtomicAdd(cntT + edst[e], 1.0f);
}

__global__ void make_inv(float* __restrict__ c, int n)
{
  int i = blockIdx.x * blockDim.x + threadIdx.x;
  if (i < n) c[i] = 1.0f / fmaxf(c[i], 1.0f);
}

// ---------------------------------------------------------------------------
// agg[sidx[e]][:] += h[gidx[e]][:]   (one wave per edge, float4 per lane,
// hardware FP32 atomics; accumulators are L2-resident)
// ---------------------------------------------------------------------------
__global__ __launch_bounds__(256) void scatter_add_feats(
    const float* __restrict__ h, const int* __restrict__ gidx,
    const int* __restrict__ sidx, float* __restrict__ agg, int E)
{
  int e = blockIdx.x * 8 + (threadIdx.x >> 5);
  if (e >= E) return;
  int lane = threadIdx.x & 31;
  int g = gidx[e];
  int s = sidx[e];
  const float4 v = *(const float4*)(h + (size_t)g * H + lane * 4);
  float* dst = agg + (size_t)s * H + lane * 4;
  unsafeAtomicAdd(dst + 0, v.x);
  unsafeAtomicAdd(dst + 1, v.y);
  unsafeAtomicAdd(dst + 2, v.z);
  unsafeAtomicAdd(dst + 3, v.w);
}

// ---------------------------------------------------------------------------
// Per-edge MLP: pred = relu([h_lig[src], h_tgt[dst]] @ Wp1 + bp1) @ Wp2 + bp2
// One wave = 16 edges (M dim). K = 256 (128 lig || 128 tgt), fully in WMMA.
// Hidden stays in registers; final dot + 16-lane shuffle reduction.
// ---------------------------------------------------------------------------
__global__ __launch_bounds__(256) void edge_mlp(
    const float* __restrict__ hl, const float* __restrict__ ht,
    const int* __restrict__ esrc, const int* __restrict__ edst,
    const float* __restrict__ Wp1, const float* __restrict__ bp1,
    const float* __restrict__ Wp2, const float* __restrict__ bp2,
    float* __restrict__ out, int E)
{
  __shared__ float ldsWT[H * LDS_P];
  const int tid  = threadIdx.x;
  const int wave = tid >> 5;
  const int lane = tid & 31;
  const int m    = lane & 15;
  const int koff = (lane < 16) ? 0 : 2;
  const int ebase = blockIdx.x * 128 + wave * 16;

  int e  = ebase + m;
  int ec = (e < E) ? e : (E - 1);
  const float* rowL = hl + (size_t)esrc[ec] * H;
  const float* rowT = ht + (size_t)edst[ec] * H;

  v8f acc[8] = {};
  for (int kc = 0; kc < 2 * H; kc += 32) {
    for (int idx = tid; idx < 32 * H; idx += 256) {
      int kl = idx >> 7;
      int n  = idx & (H - 1);
      ldsWT[n * LDS_P + kl] = Wp1[(size_t)(kc + kl) * H + n];
    }
    __syncthreads();
    for (int ks = 0; ks < 8; ++ks) {
      int kg = kc + ks * 4 + koff;
      const float* src = (kg < H) ? (rowL + kg) : (rowT + (kg - H));
      v2f a = *(const v2f*)src;           // gathered A fragment (k-contiguous)
      #pragma unroll
      for (int ct = 0; ct < 8; ++ct) {
        v2f b = *(const v2f*)&ldsWT[(ct * 16 + m) * LDS_P + ks * 4 + koff];
        acc[ct] = wmma_f32_k4(a, b, acc[ct]);
      }
    }
    __syncthreads();
  }

  // relu(hidden + bp1) dotted with Wp2, reduced over N (16 lanes per half)
  float partial[8] = {};
  #pragma unroll
  for (int ct = 0; ct < 8; ++ct) {
    int ncol = ct * 16 + m;
    float bb = bp1[ncol];
    float w2 = Wp2[ncol];
    #pragma unroll
    for (int v = 0; v < 8; ++v) {
      float hv = fmaxf(acc[ct][v] + bb, 0.0f);
      partial[v] += hv * w2;
    }
  }
  #pragma unroll
  for (int v = 0; v < 8; ++v) {
    #pragma unroll
    for (int off = 1; off < 16; off <<= 1)
      partial[v] += __shfl_xor(partial[v], off, 16);
  }
  if (m == 0) {
    float b2 = bp2[0];
    int mbase = (lane < 16) ? 0 : 8;       // lane0 -> rows 0..7, lane16 -> rows 8..15
    #pragma unroll
    for (int v = 0; v < 8; ++v) {
      int ew = ebase + mbase + v;
      if (ew < E) out[ew] = partial[v] + b2;
    }
  }
}

// ---------------------------------------------------------------------------
extern "C" void kernel_launch(void* const* d_in, const int* in_sizes, int n_in,
                              void* d_out, int out_size, void* d_ws, size_t ws_size,
                              hipStream_t stream) {
  const float* x_lig = (const float*)d_in[0];
  const float* x_tgt = (const float*)d_in[1];
  const int*   esrc  = (const int*)d_in[2];
  const int*   edst  = (const int*)d_in[3];
  const float* W_lig = (const float*)d_in[4];  const float* b_lig = (const float*)d_in[5];
  const float* W_tgt = (const float*)d_in[6];  const float* b_tgt = (const float*)d_in[7];
  const float* W1f = (const float*)d_in[8];    const float* b1f = (const float*)d_in[9];
  const float* W1r = (const float*)d_in[10];   const float* b1r = (const float*)d_in[11];
  const float* W2f = (const float*)d_in[12];   const float* b2f = (const float*)d_in[13];
  const float* W2r = (const float*)d_in[14];   const float* b2r = (const float*)d_in[15];
  const float* Wp1 = (const float*)d_in[16];   const float* bp1 = (const float*)d_in[17];
  const float* Wp2 = (const float*)d_in[18];   const float* bp2 = (const float*)d_in[19];
  float* out = (float*)d_out;

  const int NL = 100000, NT = 20000;
  const int E = in_sizes[2];

  // workspace layout (fp32): ~185 MB total
  float* p   = (float*)d_ws;
  float* hl0 = p; p += (size_t)NL * H;
  float* hl1 = p; p += (size_t)NL * H;
  float* ht0 = p; p += (size_t)NT * H;
  float* ht1 = p; p += (size_t)NT * H;
  float* aggL = p; p += (size_t)NL * H;   // aggL/aggT contiguous (single memset)
  float* aggT = p; p += (size_t)NT * H;
  float* invL = p; p += NL;               // invL/invT contiguous
  float* invT = p; p += NT;

  // degree counts -> reciprocals (edge structure identical for both convs)
  hipMemsetAsync(invL, 0, (size_t)(NL + NT) * sizeof(float), stream);
  count_edges<<<(E + 255) / 256, 256, 0, stream>>>(esrc, edst, invL, invT, E);
  make_inv<<<(NL + NT + 255) / 256, 256, 0, stream>>>(invL, NL + NT);

  // node embeddings
  gemm_bias_act<<<(NL + 63) / 64, 256, 0, stream>>>(x_lig, W_lig, b_lig, nullptr, hl0, NL, 4,    0);
  gemm_bias_act<<<(NT + 63) / 64, 256, 0, stream>>>(x_tgt, W_tgt, b_tgt, nullptr, ht0, NT, 1280, 0);

  // conv layer 1
  hipMemsetAsync(aggL, 0, (size_t)(NL + NT) * H * sizeof(float), stream);
  scatter_add_feats<<<(E + 7) / 8, 256, 0, stream>>>(hl0, esrc, edst, aggT, E);
  scatter_add_feats<<<(E + 7) / 8, 256, 0, stream>>>(ht0, edst, esrc, aggL, E);
  gemm_bias_act<<<(NT + 63) / 64, 256, 0, stream>>>(aggT, W1f, b1f, invT, ht1, NT, H, 1);
  gemm_bias_act<<<(NL + 63) / 64, 256, 0, stream>>>(aggL, W1r, b1r, invL, hl1, NL, H, 1);

  // conv layer 2
  hipMemsetAsync(aggL, 0, (size_t)(NL + NT) * H * sizeof(float), stream);
  scatter_add_feats<<<(E + 7) / 8, 256, 0, stream>>>(hl1, esrc, edst, aggT, E);
  scatter_add_feats<<<(E + 7) / 8, 256, 0, stream>>>(ht1, edst, esrc, aggL, E);
  gemm_bias_act<<<(NT + 63) / 64, 256, 0, stream>>>(aggT, W2f, b2f, invT, ht0, NT, H, 1);
  gemm_bias_act<<<(NL + 63) / 64, 256, 0, stream>>>(aggL, W2r, b2r, invL, hl0, NL, H, 1);

  // per-edge MLP
  edge_mlp<<<(E + 127) / 128, 256, 0, stream>>>(hl0, ht0, esrc, edst,
                                                Wp1, bp1, Wp2, bp2, out, E);
}